// Decoder_30305289241097
// MI455X (gfx1250) — compile-verified
//
#include <hip/hip_runtime.h>
#include <hip/hip_bf16.h>
#include <math.h>

// Problem dimensions (match reference)
#define B_   512     // batch
#define T_   50      // frames
#define F_   1128    // feature dim
#define E_   128     // embedding dim
#define H_   128     // hidden dim
#define A_   128     // attention dim
#define V_   3056    // vocab
#define L_   30      // max caption len
#define G_   512     // 4*H (gates)
#define XC   1256    // E + F (lstm concat input length)
#define XHP  1408    // padded K for [x | h]: 1256 + 128 + 24 pad (divisible by 32)
#define KC   32      // K chunk per LDS stage
#define KCP  36      // padded LDS row stride (floats) — avoids 32-bank conflicts

typedef __attribute__((ext_vector_type(2))) float v2f;
typedef __attribute__((ext_vector_type(4))) float v4f;
typedef __attribute__((ext_vector_type(8))) float v8f;
typedef int v4i_g __attribute__((vector_size(16)));   // matches builtin's int4

// CDNA5 async global->LDS path (guarded: falls back to sync copy if missing)
#if defined(__has_builtin)
#if __has_builtin(__builtin_amdgcn_global_load_async_to_lds_b128) && \
    __has_builtin(__builtin_amdgcn_s_wait_asynccnt)
#define USE_ASYNC_LDS 1
#endif
#endif

typedef __attribute__((address_space(1))) v4i_g* as1_v4i_ptr;
typedef __attribute__((address_space(3))) v4i_g* as3_v4i_ptr;

// D = A(16x4,f32) * B(4x16,f32) + C(16x16,f32)   — native fp32 WMMA
__device__ __forceinline__ v8f wmma4(v2f a, v2f b, v8f c) {
  return __builtin_amdgcn_wmma_f32_16x16x4_f32(
      /*neg_a=*/false, a, /*neg_b=*/false, b,
      /*c_mod=*/(short)0, c, /*reuse_a=*/false, /*reuse_b=*/false);
}

// -------------------------------------------------------------------------
// init: zero t=0 output slabs, c, the h+pad region of xh; tok = <sos> = 1
// -------------------------------------------------------------------------
__global__ void init_kernel(float* __restrict__ sent0, float* __restrict__ hid0,
                            float* __restrict__ xh, float* __restrict__ c,
                            int* __restrict__ tok) {
  int i = blockIdx.x * blockDim.x + threadIdx.x;
  int stride = gridDim.x * blockDim.x;
  for (int j = i; j < B_ * V_; j += stride) sent0[j] = 0.0f;
  for (int j = i; j < B_ * H_; j += stride) { hid0[j] = 0.0f; c[j] = 0.0f; }
  // xh[:, XC : XHP] = 0   (h state + K padding)
  for (int j = i; j < B_ * (XHP - XC); j += stride) {
    int b = j / (XHP - XC), k = j % (XHP - XC);
    xh[(size_t)b * XHP + XC + k] = 0.0f;
  }
  for (int j = i; j < B_; j += stride) tok[j] = 1;
}

// -------------------------------------------------------------------------
// prep: Wcat = [W_ih | W_hh | 0]  (G x XHP), biascat = b_ih + b_hh
// -------------------------------------------------------------------------
__global__ void prep_kernel(const float* __restrict__ W_ih,
                            const float* __restrict__ W_hh,
                            const float* __restrict__ b_ih,
                            const float* __restrict__ b_hh,
                            float* __restrict__ Wcat,
                            float* __restrict__ biascat) {
  int i = blockIdx.x * blockDim.x + threadIdx.x;
  int stride = gridDim.x * blockDim.x;
  for (int j = i; j < G_ * XHP; j += stride) {
    int g = j / XHP, k = j % XHP;
    float v = 0.0f;
    if (k < XC)            v = W_ih[(size_t)g * XC + k];
    else if (k < XC + H_)  v = W_hh[(size_t)g * H_ + (k - XC)];
    Wcat[j] = v;
  }
  for (int j = i; j < G_; j += stride) biascat[j] = b_ih[j] + b_hh[j];
}

// -------------------------------------------------------------------------
// Blocked WMMA GEMM:  D[M,N] = A[M,K] @ B[N,K]^T + bias[N]
// Block: 128 threads (4 waves). Block tile 64(M) x 64(N); wave w owns M-band
// w, register-blocks 4 N-tiles (4 accumulators). B K-slab (64 x KC) staged in
// LDS (double-buffered) via async global->LDS DMA, shared by all 4 waves.
// N tail handled BRANCHLESSLY: loader clamps the B row index (duplicate valid
// rows), all 4 tiles are computed unconditionally, only stores are guarded.
// K must be a multiple of KC; M a multiple of 64.
// -------------------------------------------------------------------------
__global__ __launch_bounds__(128) void gemm_kernel(
    const float* __restrict__ A, int lda,
    const float* __restrict__ Bm, int ldb,
    const float* __restrict__ bias,
    float* __restrict__ D, int ldd,
    int N, int K) {
  __shared__ float bs[2][64 * KCP];

  const int tid  = threadIdx.x;
  const int wave = tid >> 5;
  const int lane = tid & 31;
  const int lo   = lane & 15;
  const int grp  = lane >> 4;            // A frag: K = 2*grp + {0,1}
  const int nb   = blockIdx.x * 64;      // N block start
  const int mb   = blockIdx.y * 64 + wave * 16;
  const int nrem = (N - nb) >> 4;
  const int nvalid = nrem < 4 ? nrem : 4;  // valid 16-wide N tiles (epilogue only)
  const int NC = K / KC;

  const float* abase = A + (size_t)(mb + lo) * lda;

  // Cooperative B-slab stage: 64 rows x KC floats = 512 float4, 4 per thread.
  // Tail rows clamp to N-1 (in-bounds duplicates; results discarded at store).
  auto issueB = [&](int buf, int kbase) {
    #pragma unroll
    for (int q = 0; q < 4; ++q) {
      const int idx = tid + q * 128;
      const int n   = idx >> 3;          // 0..63
      const int kq  = (idx & 7) << 2;    // 0,4,...,28
      int row = nb + n;
      row = row < (N - 1) ? row : (N - 1);
      const float* g = Bm + (size_t)row * ldb + kbase + kq;
      float* l = &bs[buf][n * KCP + kq];
#if defined(USE_ASYNC_LDS)
      __builtin_amdgcn_global_load_async_to_lds_b128(
          (as1_v4i_ptr)g, (as3_v4i_ptr)l, 0, 0);
#else
      *(v4f*)l = *(const v4f*)g;
#endif
    }
  };

  v8f acc0 = {}, acc1 = {}, acc2 = {}, acc3 = {};

  int buf = 0;
  issueB(0, 0);
  for (int i = 0; i < NC; ++i) {
    __syncthreads();                       // previous chunk's buffer is free
    if (i + 1 < NC) {
      issueB(buf ^ 1, (i + 1) * KC);
#if defined(USE_ASYNC_LDS)
      __builtin_amdgcn_s_wait_asynccnt(4); // chunk i arrived (in-order)
#endif
      __builtin_prefetch(abase + (size_t)(i + 1) * KC, 0, 3);
    } else {
#if defined(USE_ASYNC_LDS)
      __builtin_amdgcn_s_wait_asynccnt(0);
#endif
    }
    __syncthreads();                       // chunk i visible to all waves

    // Batch-load this chunk's 8 A fragments (one clause of global_load_b64).
    v2f af[8];
    #pragma unroll
    for (int kk8 = 0; kk8 < 8; ++kk8)
      af[kk8] = *(const v2f*)(abase + i * KC + kk8 * 4 + (grp << 1));

    const float* bsl = bs[buf];
    #pragma unroll
    for (int kk8 = 0; kk8 < 8; ++kk8) {
      const int boff = kk8 * 4 + (grp << 1);
      acc0 = wmma4(af[kk8], *(const v2f*)(bsl + (0 * 16 + lo) * KCP + boff), acc0);
      acc1 = wmma4(af[kk8], *(const v2f*)(bsl + (1 * 16 + lo) * KCP + boff), acc1);
      acc2 = wmma4(af[kk8], *(const v2f*)(bsl + (2 * 16 + lo) * KCP + boff), acc2);
      acc3 = wmma4(af[kk8], *(const v2f*)(bsl + (3 * 16 + lo) * KCP + boff), acc3);
    }
    buf ^= 1;
  }

  v8f accs[4] = {acc0, acc1, acc2, acc3};
  #pragma unroll
  for (int j = 0; j < 4; ++j) {
    if (j < nvalid) {
      const int col = nb + j * 16 + lo;
      const float bv = bias[col];
      #pragma unroll
      for (int r = 0; r < 8; ++r)
        D[(size_t)(mb + r + 8 * grp) * ldd + col] = accs[j][r] + bv;
    }
  }
}

// -------------------------------------------------------------------------
// att_f[b,t,a] = sum_f features[b,t,f] * Wf[a,f] + bf[a]   (one-time GEMM)
// M = B*T = 25600, N = 128, K = 1128. One wave per 16x16 tile; direct loads
// (features stream is HBM-bound once; K=1128 not divisible by KC).
// -------------------------------------------------------------------------
__global__ __launch_bounds__(32) void attf_kernel(
    const float* __restrict__ features,  // [B*T, F]
    const float* __restrict__ Wf,        // [A, F]
    const float* __restrict__ bf,        // [A]
    float* __restrict__ att_f)           // [B*T, A]
{
  const int nt = blockIdx.x % (A_ / 16);
  const int mt = blockIdx.x / (A_ / 16);
  const int lane = threadIdx.x;
  const int lo = lane & 15;
  const int grp = lane >> 4;

  const float* arow = features + (size_t)(mt * 16 + lo) * F_;
  const float* brow = Wf       + (size_t)(nt * 16 + lo) * F_;

  v8f acc = {};
  #pragma unroll 2
  for (int k = 0; k < F_; k += 4) {
    v2f a = *(const v2f*)(arow + k + (grp << 1));
    v2f b = *(const v2f*)(brow + k + (grp << 1));
    acc = wmma4(a, b, acc);
  }
  const float bias = bf[nt * 16 + lo];
  #pragma unroll
  for (int r = 0; r < 8; ++r) {
    const int m = mt * 16 + r + 8 * grp;
    att_f[(size_t)m * A_ + nt * 16 + lo] = acc[r] + bias;
  }
}

// -------------------------------------------------------------------------
// attention + context + embedding gather. One 128-thread block per batch row.
// Reads h from xh[:, XC:XC+H); writes xh[:, 0:XC) = [emb | ctx].
// -------------------------------------------------------------------------
__global__ __launch_bounds__(128) void attention_kernel(
    const float* __restrict__ features,   // [B,T,F]
    const float* __restrict__ embedding,  // [V,E]
    const float* __restrict__ Wh,         // [A,H]
    const float* __restrict__ bh,         // [A]
    const float* __restrict__ wa,         // [A]
    const float* __restrict__ ba,         // [1]
    const float* __restrict__ att_f,      // [B,T,A]
    const int*   __restrict__ tok,        // [B]
    float* __restrict__ xh)               // [B, XHP]
{
  const int b = blockIdx.x;
  const int tid = threadIdx.x;
  __shared__ float h_s[H_], ah_s[A_], al_s[T_], wa_s[A_];

  h_s[tid]  = xh[(size_t)b * XHP + XC + tid];
  wa_s[tid] = wa[tid];
  __syncthreads();

  // att_h[a] = h . Wh[a,:] + bh[a]
  float acc = bh[tid];
  const float* wr = Wh + (size_t)tid * H_;
  #pragma unroll 8
  for (int k = 0; k < H_; ++k) acc += h_s[k] * wr[k];
  ah_s[tid] = acc;
  __syncthreads();

  // e[t] = tanh(att_h + att_f[b,t]) . wa + ba
  if (tid < T_) {
    const float* af = att_f + ((size_t)b * T_ + tid) * A_;
    float e = ba[0];
    for (int a = 0; a < A_; ++a) e += tanhf(ah_s[a] + af[a]) * wa_s[a];
    al_s[tid] = e;
  }
  __syncthreads();

  if (tid == 0) {  // softmax over T (tiny)
    float mx = al_s[0];
    for (int t = 1; t < T_; ++t) mx = fmaxf(mx, al_s[t]);
    float s = 0.0f;
    for (int t = 0; t < T_; ++t) { al_s[t] = expf(al_s[t] - mx); s += al_s[t]; }
    const float inv = 1.0f / s;
    for (int t = 0; t < T_; ++t) al_s[t] *= inv;
  }
  __syncthreads();

  // embedding gather -> xh[:, 0:E)
  xh[(size_t)b * XHP + tid] = embedding[(size_t)tok[b] * E_ + tid];

  // ctx -> xh[:, E:XC)   (features stays L2-resident: 115MB < 192MB L2)
  const float* fb = features + (size_t)b * T_ * F_;
  for (int f = tid; f < F_; f += 128) {
    float s = 0.0f;
    for (int t = 0; t < T_; ++t) s += al_s[t] * fb[(size_t)t * F_ + f];
    xh[(size_t)b * XHP + E_ + f] = s;
  }
}

// -------------------------------------------------------------------------
// LSTM pointwise (PyTorch gate order i,f,g,o); h lives inside xh.
// -------------------------------------------------------------------------
__global__ __launch_bounds__(256) void lstm_pointwise_kernel(
    const float* __restrict__ gates, float* __restrict__ xh,
    float* __restrict__ c, float* __restrict__ hid_out) {
  const int tid = blockIdx.x * blockDim.x + threadIdx.x;   // B*H total
  const int b = tid / H_, j = tid % H_;
  const float* g = gates + (size_t)b * G_;
  const float i_ = g[j], f_ = g[H_ + j], g_ = g[2 * H_ + j], o_ = g[3 * H_ + j];
  const float si = 1.0f / (1.0f + expf(-i_));
  const float sf = 1.0f / (1.0f + expf(-f_));
  const float so = 1.0f / (1.0f + expf(-o_));
  const float cn = sf * c[tid] + si * tanhf(g_);
  const float hn = so * tanhf(cn);
  c[tid] = cn;
  xh[(size_t)b * XHP + XC + j] = hn;
  hid_out[tid] = hn;
}

// -------------------------------------------------------------------------
// In-place log_softmax over V + first-index argmax -> tok. One block per row.
// -------------------------------------------------------------------------
__global__ __launch_bounds__(256) void lsm_argmax_kernel(
    float* __restrict__ row_base, int* __restrict__ tok) {
  const int b = blockIdx.x;
  float* row = row_base + (size_t)b * V_;
  __shared__ float smax[256]; __shared__ int sidx[256]; __shared__ float ssum[256];
  const int tid = threadIdx.x;

  float mx = -INFINITY; int mi = V_;
  for (int v = tid; v < V_; v += 256) {
    const float val = row[v];
    if (val > mx) { mx = val; mi = v; }
  }
  smax[tid] = mx; sidx[tid] = mi; __syncthreads();
  for (int s = 128; s > 0; s >>= 1) {
    if (tid < s) {
      if (smax[tid + s] > smax[tid] ||
          (smax[tid + s] == smax[tid] && sidx[tid + s] < sidx[tid])) {
        smax[tid] = smax[tid + s]; sidx[tid] = sidx[tid + s];
      }
    }
    __syncthreads();
  }
  const float gmax = smax[0];

  float sum = 0.0f;
  for (int v = tid; v < V_; v += 256) sum += expf(row[v] - gmax);
  ssum[tid] = sum; __syncthreads();
  for (int s = 128; s > 0; s >>= 1) {
    if (tid < s) ssum[tid] += ssum[tid + s];
    __syncthreads();
  }
  const float lse = gmax + logf(ssum[0]);

  for (int v = tid; v < V_; v += 256) row[v] -= lse;
  if (tid == 0) tok[b] = sidx[0];
}

// -------------------------------------------------------------------------
extern "C" void kernel_launch(void* const* d_in, const int* in_sizes, int n_in,
                              void* d_out, int out_size, void* d_ws, size_t ws_size,
                              hipStream_t stream) {
  const float* features  = (const float*)d_in[0];
  const float* embedding = (const float*)d_in[1];
  const float* Wh    = (const float*)d_in[2];
  const float* bh    = (const float*)d_in[3];
  const float* Wf    = (const float*)d_in[4];
  const float* bf    = (const float*)d_in[5];
  const float* wa    = (const float*)d_in[6];
  const float* ba    = (const float*)d_in[7];
  const float* W_ih  = (const float*)d_in[8];
  const float* W_hh  = (const float*)d_in[9];
  const float* b_ih  = (const float*)d_in[10];
  const float* b_hh  = (const float*)d_in[11];
  const float* W_out = (const float*)d_in[12];
  const float* b_out = (const float*)d_in[13];

  float* out  = (float*)d_out;
  float* sent = out;                           // [L, B, V]
  float* hid  = out + (size_t)L_ * B_ * V_;    // [L, 1, B, H]

  // workspace carve-out (~20.2 MB)
  float* ws      = (float*)d_ws;
  float* att_f   = ws;  ws += (size_t)B_ * T_ * A_;   // 3,276,800
  float* xh      = ws;  ws += (size_t)B_ * XHP;       //   720,896 ([x | h | pad])
  float* c       = ws;  ws += (size_t)B_ * H_;        //    65,536
  float* gates   = ws;  ws += (size_t)B_ * G_;        //   262,144
  float* Wcat    = ws;  ws += (size_t)G_ * XHP;       //   720,896
  float* biascat = ws;  ws += G_;                     //       512
  int*   tok     = (int*)ws;                          //       512

  init_kernel<<<256, 256, 0, stream>>>(sent, hid, xh, c, tok);
  prep_kernel<<<512, 256, 0, stream>>>(W_ih, W_hh, b_ih, b_hh, Wcat, biascat);

  // one-time attention feature projection (WMMA GEMM, 3.7G MACs)
  attf_kernel<<<(B_ * T_ / 16) * (A_ / 16), 32, 0, stream>>>(features, Wf, bf, att_f);

  for (int s = 0; s < L_ - 1; ++s) {
    const int t = s + 1;
    attention_kernel<<<B_, 128, 0, stream>>>(features, embedding, Wh, bh, wa, ba,
                                             att_f, tok, xh);
    // gates = [x|h] @ Wcat^T + biascat : M=512, N=512, K=1408
    gemm_kernel<<<dim3(G_ / 64, B_ / 64), 128, 0, stream>>>(
        xh, XHP, Wcat, XHP, biascat, gates, G_, G_, XHP);
    lstm_pointwise_kernel<<<B_ * H_ / 256, 256, 0, stream>>>(
        gates, xh, c, hid + (size_t)t * B_ * H_);
    // logits = h @ W_out^T + b_out : M=512, N=3056, K=128
    gemm_kernel<<<dim3((V_ + 63) / 64, B_ / 64), 128, 0, stream>>>(
        xh + XC, XHP, W_out, H_, b_out, sent + (size_t)t * B_ * V_, V_, V_, H_);
    lsm_argmax_kernel<<<B_, 256, 0, stream>>>(sent + (size_t)t * B_ * V_, tok);
  }
}